// GAT_linear_29832842838723
// MI455X (gfx1250) — compile-verified
//
#include <hip/hip_runtime.h>
#include <math.h>

typedef float v2f __attribute__((ext_vector_type(2)));
typedef float v8f __attribute__((ext_vector_type(8)));

#define IN_F  128
#define HID_F 48
#define OUT_F 32
#define NEG_SLOPE 0.2f
#define MAX_B_ELEMS (IN_F * HID_F)   // largest weight matrix: 128*48 = 6144 f32 = 24 KB LDS

// ---------------------------------------------------------------------------
// f32 WMMA GEMM: C[M][Nc] = A[M][K] * B[K][Nc] (+ bias[Nc] if given)
// One wave32 computes one 16x16 tile of C using V_WMMA_F32_16X16X4_F32.
// The full weight matrix B (<= 24KB) is staged into LDS once per workgroup;
// WMMA B-fragments then come from ds_load while A streams from global (b64).
// A-frag (16x4 f32): VGPR0 = K=(k0 + 2*half), VGPR1 = K=(k0 + 2*half + 1),
//   rows M = lane&15 (ISA 7.12.2).  B-frag (4x16) mirrored across lanes.
// C/D (16x16 f32, 8 VGPRs): vgpr v -> row M = v + 8*half, col N = lane&15.
// ---------------------------------------------------------------------------
__global__ void gat_gemm_wmma_f32(const float* __restrict__ A,
                                  const float* __restrict__ B,
                                  const float* __restrict__ bias,
                                  float* __restrict__ C,
                                  int M, int K, int Nc) {
  __shared__ float Bs[MAX_B_ELEMS];
  const int nB = K * Nc;
  for (int i = (int)threadIdx.x; i < nB; i += (int)blockDim.x) Bs[i] = B[i];
  __syncthreads();

  const int wave = (int)((blockIdx.x * blockDim.x + threadIdx.x) >> 5);
  const int lane = (int)(threadIdx.x & 31);
  const int tilesN = Nc >> 4;
  const int tm = wave / tilesN;
  const int tn = wave % tilesN;
  if (tm * 16 >= M) return;

  const int m0 = tm * 16;
  const int n0 = tn * 16;
  const int half = lane >> 4;   // 0: lanes 0-15, 1: lanes 16-31
  const int l15  = lane & 15;

  const float* __restrict__ arow = A + (size_t)(m0 + l15) * K;
  const float* __restrict__ bcol = Bs + n0 + l15;
  v8f acc = {};
  for (int k0 = 0; k0 < K; k0 += 4) {
    const int ka = k0 + half * 2;           // even -> 8B aligned
    v2f a = *(const v2f*)(arow + ka);       // global_load_b64
    v2f b;
    b.x = bcol[ka * Nc];                    // ds_load
    b.y = bcol[(ka + 1) * Nc];
    acc = __builtin_amdgcn_wmma_f32_16x16x4_f32(
        /*neg_a=*/false, a, /*neg_b=*/false, b,
        /*c_mod=*/(short)0, acc, /*reuse_a=*/false, /*reuse_b=*/false);
  }

  const float bv = bias ? bias[n0 + l15] : 0.0f;
#pragma unroll
  for (int v = 0; v < 8; ++v) {
    const int row = m0 + v + half * 8;
    C[(size_t)row * Nc + n0 + l15] = acc[v] + bv;
  }
}

// ---------------------------------------------------------------------------
// Per-node attention scalars: asrc[i] = h[i].a_src ; adst[i] = h[i].a_dst
// ---------------------------------------------------------------------------
__global__ void gat_node_alpha(const float* __restrict__ h,
                               const float* __restrict__ a_src,
                               const float* __restrict__ a_dst,
                               float* __restrict__ asrc,
                               float* __restrict__ adst, int N, int F) {
  const int i = (int)(blockIdx.x * blockDim.x + threadIdx.x);
  if (i >= N) return;
  const float* __restrict__ row = h + (size_t)i * F;
  float s = 0.f, d = 0.f;
  for (int f = 0; f < F; ++f) {
    const float v = row[f];
    s = fmaf(v, a_src[f], s);
    d = fmaf(v, a_dst[f], d);
  }
  asrc[i] = s;
  adst[i] = d;
}

// order-preserving float<->uint keys so segment-max can use atomicMax(u32);
// key 0 (memset init) decodes below every finite float.
__device__ __forceinline__ unsigned f2ord(float f) {
  unsigned u = __float_as_uint(f);
  return (u & 0x80000000u) ? ~u : (u | 0x80000000u);
}
__device__ __forceinline__ float ord2f(unsigned k) {
  return __uint_as_float((k & 0x80000000u) ? (k & 0x7FFFFFFFu) : ~k);
}

// pass A: e = leakyrelu(asrc[src]+adst[dst]); segment max over dst
__global__ void gat_edge_logit_max(const int* __restrict__ src,
                                   const int* __restrict__ dst,
                                   const float* __restrict__ asrc,
                                   const float* __restrict__ adst,
                                   float* __restrict__ e,
                                   unsigned* __restrict__ mkey, int E) {
  const int i = (int)(blockIdx.x * blockDim.x + threadIdx.x);
  if (i >= E) return;
  const int s = src[i], d = dst[i];
  float x = asrc[s] + adst[d];
  x = x > 0.0f ? x : NEG_SLOPE * x;
  e[i] = x;
  atomicMax(&mkey[d], f2ord(x));
}

// pass B: ex = exp(e - max[dst]) (in place); segment sum over dst
__global__ void gat_edge_exp_sum(const int* __restrict__ dst,
                                 float* __restrict__ e,
                                 const unsigned* __restrict__ mkey,
                                 float* __restrict__ ssum, int E) {
  const int i = (int)(blockIdx.x * blockDim.x + threadIdx.x);
  if (i >= E) return;
  const int d = dst[i];
  const float ex = __expf(e[i] - ord2f(mkey[d]));
  e[i] = ex;
  atomicAdd(&ssum[d], ex);
}

// pass C: alpha = ex / sum[dst]; agg[dst] += alpha * h[src]  (wave per edge)
// alpha_out may alias ex (all lanes read ex before lane0 stores).
__global__ void gat_edge_scatter(const int* __restrict__ src,
                                 const int* __restrict__ dst,
                                 const float* __restrict__ ex,
                                 const float* __restrict__ ssum,
                                 const float* __restrict__ h,
                                 float* __restrict__ agg,
                                 float* __restrict__ alpha_out, int E, int F) {
  const int wave = (int)((blockIdx.x * blockDim.x + threadIdx.x) >> 5);
  const int lane = (int)(threadIdx.x & 31);
  if (wave >= E) return;
  const int s = src[wave], d = dst[wave];
  const float al = ex[wave] / ssum[d];
  if (alpha_out && lane == 0) alpha_out[wave] = al;
  const float* __restrict__ hrow = h + (size_t)s * F;
  float* __restrict__ orow = agg + (size_t)d * F;
  for (int f = lane; f < F; f += 32) atomicAdd(&orow[f], al * hrow[f]);
}

// node activation: mode 0 -> ELU(x+b), mode 1 -> ReLU(x+b); in place
__global__ void gat_node_act(float* __restrict__ h, const float* __restrict__ b,
                             int N, int F, int mode) {
  const int i = (int)(blockIdx.x * blockDim.x + threadIdx.x);
  if (i >= N * F) return;
  const int f = i % F;
  float v = h[i] + b[f];
  if (mode == 0) v = v > 0.0f ? v : (__expf(v) - 1.0f);   // ELU
  else           v = v > 0.0f ? v : 0.0f;                 // ReLU
  h[i] = v;
}

extern "C" void kernel_launch(void* const* d_in, const int* in_sizes, int n_in,
                              void* d_out, int out_size, void* d_ws, size_t ws_size,
                              hipStream_t stream) {
  const float* x      = (const float*)d_in[0];
  const int*   eidx   = (const int*)d_in[1];
  const float* W1     = (const float*)d_in[2];
  const float* a_src1 = (const float*)d_in[3];
  const float* a_dst1 = (const float*)d_in[4];
  const float* b1     = (const float*)d_in[5];
  const float* W2     = (const float*)d_in[6];
  const float* a_src2 = (const float*)d_in[7];
  const float* a_dst2 = (const float*)d_in[8];
  const float* b2     = (const float*)d_in[9];
  const float* Wl     = (const float*)d_in[10];
  const float* bl     = (const float*)d_in[11];

  const int N = in_sizes[0] / IN_F;
  const int E = in_sizes[1] / 2;
  const int* src = eidx;
  const int* dst = eidx + E;

  // workspace layout (floats)
  float* ws   = (float*)d_ws;
  float* h1   = ws;                       // N*HID
  float* h1a  = h1  + (size_t)N * HID_F;  // N*HID (agg, zeroed)
  float* h2   = h1a + (size_t)N * HID_F;  // N*OUT
  float* h2a  = h2  + (size_t)N * OUT_F;  // N*OUT (agg, zeroed)
  float* asrc = h2a + (size_t)N * OUT_F;  // N
  float* adst = asrc + N;                 // N
  unsigned* mkey = (unsigned*)(adst + N); // N (zeroed per layer)
  float* ssum = (float*)(mkey + N);       // N (zeroed per layer)
  float* e1   = ssum + N;                 // E

  float* out_mean  = (float*)d_out;                // N*OUT
  float* out_alpha = out_mean + (size_t)N * OUT_F; // E (also layer-2 edge buf)

  const int TB = 256;
  const int nodeBlk  = (N + TB - 1) / TB;
  const int edgeBlk  = (E + TB - 1) / TB;
  const int edgeWBlk = (int)(((size_t)E * 32 + TB - 1) / TB);
  auto gemmBlk = [&](int M, int Nc) {
    const int tiles = ((M + 15) / 16) * (Nc / 16);
    return (tiles + (TB / 32) - 1) / (TB / 32);
  };

  // ---- zero-init accumulators (graph-capture safe) ----
  hipMemsetAsync(h1a, 0, (size_t)N * HID_F * sizeof(float), stream);
  hipMemsetAsync(h2a, 0, (size_t)N * OUT_F * sizeof(float), stream);
  hipMemsetAsync(mkey, 0, (size_t)N * sizeof(unsigned), stream);
  hipMemsetAsync(ssum, 0, (size_t)N * sizeof(float), stream);

  // ---- layer 1 ----
  gat_gemm_wmma_f32<<<gemmBlk(N, HID_F), TB, 0, stream>>>(x, W1, nullptr, h1, N, IN_F, HID_F);
  gat_node_alpha<<<nodeBlk, TB, 0, stream>>>(h1, a_src1, a_dst1, asrc, adst, N, HID_F);
  gat_edge_logit_max<<<edgeBlk, TB, 0, stream>>>(src, dst, asrc, adst, e1, mkey, E);
  gat_edge_exp_sum<<<edgeBlk, TB, 0, stream>>>(dst, e1, mkey, ssum, E);
  gat_edge_scatter<<<edgeWBlk, TB, 0, stream>>>(src, dst, e1, ssum, h1, h1a, nullptr, E, HID_F);
  gat_node_act<<<(N * HID_F + TB - 1) / TB, TB, 0, stream>>>(h1a, b1, N, HID_F, /*ELU*/0);

  // ---- layer 2 ----
  gat_gemm_wmma_f32<<<gemmBlk(N, OUT_F), TB, 0, stream>>>(h1a, W2, nullptr, h2, N, HID_F, OUT_F);
  gat_node_alpha<<<nodeBlk, TB, 0, stream>>>(h2, a_src2, a_dst2, asrc, adst, N, OUT_F);
  hipMemsetAsync(mkey, 0, (size_t)N * sizeof(unsigned), stream);
  hipMemsetAsync(ssum, 0, (size_t)N * sizeof(float), stream);
  gat_edge_logit_max<<<edgeBlk, TB, 0, stream>>>(src, dst, asrc, adst, out_alpha, mkey, E);
  gat_edge_exp_sum<<<edgeBlk, TB, 0, stream>>>(dst, out_alpha, mkey, ssum, E);
  gat_edge_scatter<<<edgeWBlk, TB, 0, stream>>>(src, dst, out_alpha, ssum, h2, h2a,
                                                out_alpha /*alpha output, in place*/, E, OUT_F);
  gat_node_act<<<(N * OUT_F + TB - 1) / TB, TB, 0, stream>>>(h2a, b2, N, OUT_F, /*ReLU*/1);

  // ---- head: mean = relu_h @ Wl + bl ----
  gat_gemm_wmma_f32<<<gemmBlk(N, OUT_F), TB, 0, stream>>>(h2a, Wl, bl, out_mean, N, OUT_F, OUT_F);
}